// DeepConv1d_63101659513165
// MI455X (gfx1250) — compile-verified
//
#include <hip/hip_runtime.h>
#include <hip/hip_bf16.h>

typedef __attribute__((ext_vector_type(2))) float v2f;
typedef __attribute__((ext_vector_type(8))) float v8f;

// ---------------------------------------------------------------------------
// Kernel 1: y[b,c,n] = conv_b[c] + sum_d conv_w[c,d] * interp(deep[b,d,:], n)
// plus per-(b,c) mean and scale = GAMA / (var + EPS)  (ddof=1 variance).
// One block per (b,c) channel (1024 blocks), 256 threads.
// ---------------------------------------------------------------------------
__global__ void __launch_bounds__(256)
k1_conv_stats(const float* __restrict__ deep, const float* __restrict__ conv_w,
              const float* __restrict__ conv_b, float* __restrict__ y,
              float* __restrict__ meanv, float* __restrict__ scalev) {
    __shared__ float s_cw[16];
    __shared__ float s_red[256];
    __shared__ float s_red2[256];
    const int bc  = blockIdx.x;          // 0..1023
    const int c   = bc & 63;
    const int b   = bc >> 6;
    const int tid = threadIdx.x;
    if (tid < 16) s_cw[tid] = conv_w[c * 16 + tid];
    __syncthreads();
    float cw[16];
#pragma unroll
    for (int d = 0; d < 16; ++d) cw[d] = s_cw[d];
    const float bias = conv_b[c];
    const float* dp = deep + (size_t)b * 16 * 1024;
    float* yp = y + (size_t)bc * 4096;
    float sum = 0.f, ss = 0.f;
    for (int n = tid; n < 4096; n += 256) {
        // src = (n+0.5)*(1024/4096) - 0.5, clipped to [0, 1023]
        float src = fmaf((float)n + 0.5f, 0.25f, -0.5f);
        src = fminf(fmaxf(src, 0.f), 1023.f);
        int lo = (int)src;               // src >= 0 -> trunc == floor
        int hi = min(lo + 1, 1023);
        float w = src - (float)lo;
        float acc = bias;
#pragma unroll
        for (int d = 0; d < 16; ++d) {
            float v0 = dp[d * 1024 + lo];
            float v1 = dp[d * 1024 + hi];
            acc = fmaf(cw[d], fmaf(w, v1 - v0, v0), acc);
        }
        yp[n] = acc;
        sum += acc;
        ss = fmaf(acc, acc, ss);
    }
    s_red[tid] = sum; s_red2[tid] = ss;
    __syncthreads();
    for (int off = 128; off > 0; off >>= 1) {
        if (tid < off) { s_red[tid] += s_red[tid + off]; s_red2[tid] += s_red2[tid + off]; }
        __syncthreads();
    }
    if (tid == 0) {
        float m   = s_red[0] * (1.f / 4096.f);
        float var = (s_red2[0] - 4096.f * m * m) * (1.f / 4095.f);
        meanv[bc]  = m;
        scalev[bc] = 0.5f / (var + 1e-9f);   // GAMA / (var + EPS)
    }
}

// ---------------------------------------------------------------------------
// Kernel 2: fused unfold + sech^2 attention + elementwise + GEMM via
// V_WMMA_F32_16X16X4_F32.  Grid (64, 16): blockIdx.x = 64-position L tile,
// blockIdx.y = batch.  256 threads = 8 waves; wave w owns output tile
// o in [16w, 16w+16); each wave runs 4 M-tiles (64 L positions) per k-step
// so the B fragment (fc_w, L2-resident) is reused 4x.
// ---------------------------------------------------------------------------
__global__ void __launch_bounds__(256, 1)
k2_fused(const float* __restrict__ x, const float* __restrict__ y,
         const float* __restrict__ meanv, const float* __restrict__ scalev,
         const float* __restrict__ fc_w, float* __restrict__ out) {
    extern __shared__ float smem[];
    float* s_xs    = smem;                 // 64 ch x 72 (70-wide window, padded)
    float* s_x     = s_xs + 64 * 72;       // 64 ch x 72
    float* s_dlY   = s_x  + 64 * 72;       // 64 l  x 452 (448 padded; stride%64==4)
    float* s_mean  = s_dlY + 64 * 452;     // 64
    float* s_scale = s_mean + 64;          // 64

    const int tid = threadIdx.x;
    const int b   = blockIdx.y;
    const int L0  = blockIdx.x * 64;

    if (tid < 64) {
        s_mean[tid]  = meanv[b * 64 + tid];
        s_scale[tid] = scalev[b * 64 + tid];
    }
    __syncthreads();

    // Stage reflect-padded windows of xs (normalized y) and x into LDS.
    const size_t rowbase = (size_t)b * 64 * 4096;
    for (int p = tid; p < 64 * 70; p += 256) {
        int c = p / 70, j = p - c * 70;
        int g = L0 - 3 + j;                 // global position, reflect pad=3
        g = (g < 0) ? -g : g;
        g = (g > 4095) ? (8190 - g) : g;
        size_t gi = rowbase + (size_t)c * 4096 + g;
        s_xs[c * 72 + j] = (y[gi] - s_mean[c]) * s_scale[c];
        s_x [c * 72 + j] = x[gi];
    }
    __syncthreads();

    // dlY[l][c*7+k] = sech^2(|xs[l-3+k]-xs[l]|) * alpha * x[l-3+k]
    for (int p = tid; p < 4096; p += 256) {
        int l = p >> 6, c = p & 63;
        const float* xw = s_xs + c * 72 + l;   // window index k -> pos l-3+k
        float xc = xw[3];
        float lv[7];
        float s = 0.f;
#pragma unroll
        for (int k = 0; k < 7; ++k) {
            float d = fabsf(xw[k] - xc);
            float e = __expf(-2.f * d);        // 1 - tanh^2(d) = 4e/(1+e)^2
            float u = 1.f + e;
            float ls = 4.f * e / (u * u);
            lv[k] = ls;
            s += ls;
        }
        float alpha = 1.f / s;
        const float* xv = s_x + c * 72 + l;
        float* dst = s_dlY + l * 452 + c * 7;
#pragma unroll
        for (int k = 0; k < 7; ++k) dst[k] = lv[k] * alpha * xv[k];
    }
    __syncthreads();

    // GEMM: D(16x16) += A(16x4) x B(4x16), K = 448 -> 112 steps.
    // A frag (ISA 7.12.2): lanes 0-15 hold K0/K1, lanes 16-31 hold K2/K3.
    // B/C/D: row striped across lanes; lanes 16-31 are the +2 (B) / +8 (D) rows.
    const int wv   = tid >> 5;             // wave id -> o-tile
    const int lane = tid & 31;
    const int m    = lane & 15;
    const int koff = (lane & 16) ? 2 : 0;
    const float* bp = fc_w + (size_t)(wv * 16 + m) * 448 + koff;  // B[k][o]=fc_w[o][k]
    const float* ap = s_dlY + m * 452 + koff;

    v8f acc0 = {}, acc1 = {}, acc2 = {}, acc3 = {};
#pragma unroll 4
    for (int kk = 0; kk < 112; ++kk) {
        v2f bfrag = *(const v2f*)(bp + 4 * kk);
        v2f a0 = *(const v2f*)(ap + 4 * kk);
        v2f a1 = *(const v2f*)(ap + 16 * 452 + 4 * kk);
        v2f a2 = *(const v2f*)(ap + 32 * 452 + 4 * kk);
        v2f a3 = *(const v2f*)(ap + 48 * 452 + 4 * kk);
        acc0 = __builtin_amdgcn_wmma_f32_16x16x4_f32(false, a0, false, bfrag, (short)0, acc0, false, false);
        acc1 = __builtin_amdgcn_wmma_f32_16x16x4_f32(false, a1, false, bfrag, (short)0, acc1, false, false);
        acc2 = __builtin_amdgcn_wmma_f32_16x16x4_f32(false, a2, false, bfrag, (short)0, acc2, false, false);
        acc3 = __builtin_amdgcn_wmma_f32_16x16x4_f32(false, a3, false, bfrag, (short)0, acc3, false, false);
    }

    // Store: D vgpr r -> M = r (+8 for lanes 16-31), N = lane&15.
    const int o   = wv * 16 + m;
    const int mhi = (lane & 16) ? 8 : 0;
    float* op = out + ((size_t)b * 128 + o) * 4096 + L0 + mhi;
    v8f accs[4] = {acc0, acc1, acc2, acc3};
#pragma unroll
    for (int mt = 0; mt < 4; ++mt) {
#pragma unroll
        for (int r = 0; r < 8; ++r) {
            op[mt * 16 + r] = accs[mt][r];
        }
    }
}

// ---------------------------------------------------------------------------
extern "C" void kernel_launch(void* const* d_in, const int* in_sizes, int n_in,
                              void* d_out, int out_size, void* d_ws, size_t ws_size,
                              hipStream_t stream) {
    (void)in_sizes; (void)n_in; (void)out_size; (void)ws_size;
    const float* deep   = (const float*)d_in[0];
    const float* x      = (const float*)d_in[1];
    const float* conv_w = (const float*)d_in[2];
    const float* conv_b = (const float*)d_in[3];
    const float* fc_w   = (const float*)d_in[4];
    float* out = (float*)d_out;

    float* ws_y     = (float*)d_ws;                       // 16*64*4096 f32 = 16 MB
    float* ws_mean  = ws_y + (size_t)16 * 64 * 4096;      // 1024 f32
    float* ws_scale = ws_mean + 1024;                     // 1024 f32

    k1_conv_stats<<<1024, 256, 0, stream>>>(deep, conv_w, conv_b, ws_y, ws_mean, ws_scale);

    // dynamic LDS: (2*64*72 + 64*452 + 128) floats = 153088 bytes (< 320KB/WGP)
    size_t lds = (size_t)(2 * 64 * 72 + 64 * 452 + 128) * sizeof(float);
    k2_fused<<<dim3(64, 16), 256, lds, stream>>>(x, ws_y, ws_mean, ws_scale, fc_w, out);
}